// MagneticEdgeConv_26233660244747
// MI455X (gfx1250) — compile-verified
//
#include <hip/hip_runtime.h>

#define NUM_NODES 2048
#define NUM_EDGES 8192
#define IN_CH     32
#define OUT_CH    16

// pi*q = pi/4  ->  cos = sin = 1/sqrt(2); also rsqrt(deg=2) = 1/sqrt(2)
__device__ __constant__ const float INV_SQRT2 = 0.70710678118654752440f;

typedef float v2f __attribute__((ext_vector_type(2)));
typedef float v8f __attribute__((ext_vector_type(8)));

// ---------------------------------------------------------------- zero node accumulator
__global__ __launch_bounds__(256) void MEC_zero(float* __restrict__ z) {
    int i = blockIdx.x * 256 + threadIdx.x;
    if (i < NUM_NODES * OUT_CH) z[i] = 0.0f;
}

// ---------------------------------------------------------------- phase 1:
// per 16-edge tile (one wave): x16 = edge_attr@W_all^T, skip16 = edge_attr@W_skip^T
// via chained V_WMMA_F32_16X16X4_F32; write skip+bias to out; scatter
// rotated+scaled complex contributions into node buffer z with f32 atomics.
__global__ __launch_bounds__(256) void MEC_wmma_scatter(
    const int*   __restrict__ ei,     // [2, E]
    const float* __restrict__ ea,     // [E, 32]
    const int*   __restrict__ und,    // [E]
    const float* __restrict__ Wall,   // [16, 32]
    const float* __restrict__ Wskip,  // [16, 32]
    const float* __restrict__ bias,   // [16]
    float*       __restrict__ out,    // [E, 16]
    float*       __restrict__ z)      // [N, 16] (8 complex per node)
{
    const int lane = threadIdx.x & 31;
    const int wave = threadIdx.x >> 5;
    const int tile = blockIdx.x * 8 + wave;          // 512 tiles of 16 edges
    const int half = lane >> 4;                      // 0: K{0,1}, 1: K{2,3}
    const int n    = lane & 15;                      // A row / B col / D col
    const int koff = half * 2;

    const float* arow = ea    + (tile * 16 + n) * IN_CH + koff;  // A: M = n
    const float* wa   = Wall  + n * IN_CH + koff;                // B: N = n (col per lane)
    const float* ws   = Wskip + n * IN_CH + koff;

    v8f acc_all = {};
    v8f acc_sk  = {};
#pragma unroll
    for (int b = 0; b < 8; ++b) {                    // K = 32 = 8 x 4
        v2f a  = *(const v2f*)(arow + 4 * b);
        v2f bw = *(const v2f*)(wa   + 4 * b);
        v2f bs = *(const v2f*)(ws   + 4 * b);
        acc_all = __builtin_amdgcn_wmma_f32_16x16x4_f32(
            false, a, false, bw, (short)0, acc_all, false, false);
        acc_sk  = __builtin_amdgcn_wmma_f32_16x16x4_f32(
            false, a, false, bs, (short)0, acc_sk, false, false);
    }

    const float bn   = bias[n];
    const int   comp = n & 1;                        // 0 = Re lane, 1 = Im lane
#pragma unroll
    for (int r = 0; r < 8; ++r) {
        const int e = tile * 16 + r + 8 * half;      // D row -> edge id
        // skip connection + bias (Laplacian part added in phase 2)
        out[e * OUT_CH + n] = acc_sk[r] + bn;

        const int s = ei[e];
        const int d = ei[NUM_EDGES + e];
        const int u = und[e];
        const float cs   = u ? 1.0f : INV_SQRT2;     // cos(theta)
        const float sn   = u ? 0.0f : INV_SQRT2;     // sin(theta)
        const float invs = (s == d && u) ? 0.0f : INV_SQRT2;  // deg in {0,2}

        const float uval = invs * acc_all[r];        // D^{-1/2} x (one component)
        const float pval = __shfl_xor(uval, 1, 32);  // complex partner
        const float re   = comp ? pval : uval;
        const float im   = comp ? uval : pval;
        //   z[src] +=  sig * u          (sig = cs + i*sn)
        //   z[dst] += -conj(sig) * u    (s = -1, orientation-equivariant)
        const float vsrc = comp ? ( sn * re + cs * im) : ( cs * re - sn * im);
        const float vdst = comp ? ( sn * re - cs * im) : (-cs * re - sn * im);
        atomicAdd(&z[s * OUT_CH + n], vsrc);
        atomicAdd(&z[d * OUT_CH + n], vdst);
    }
}

// ---------------------------------------------------------------- phase 2:
// per (edge, channel): w = invs * (conj(sig)*z[src] - sig*z[dst]); out += w
__global__ __launch_bounds__(256) void MEC_gather(
    const int*   __restrict__ ei,
    const int*   __restrict__ und,
    const float* __restrict__ z,
    float*       __restrict__ out)
{
    const int tid  = blockIdx.x * 256 + threadIdx.x;   // E*16 threads
    const int e    = tid >> 4;
    const int n    = tid & 15;
    const int c    = n >> 1;
    const int comp = n & 1;

    const int s = ei[e];
    const int d = ei[NUM_EDGES + e];
    const int u = und[e];
    const float cs   = u ? 1.0f : INV_SQRT2;
    const float sn   = u ? 0.0f : INV_SQRT2;
    const float invs = (s == d && u) ? 0.0f : INV_SQRT2;

    const float2 zs = ((const float2*)z)[s * (OUT_CH / 2) + c];
    const float2 zd = ((const float2*)z)[d * (OUT_CH / 2) + c];

    // conj(sig)*zs - sig*zd:
    //   Re = cs*(zs.x - zd.x) + sn*(zs.y + zd.y)
    //   Im = cs*(zs.y - zd.y) - sn*(zs.x + zd.x)
    float w;
    if (comp == 0) w = cs * (zs.x - zd.x) + sn * (zs.y + zd.y);
    else           w = cs * (zs.y - zd.y) - sn * (zs.x + zd.x);

    out[tid] += invs * w;
}

extern "C" void kernel_launch(void* const* d_in, const int* in_sizes, int n_in,
                              void* d_out, int out_size, void* d_ws, size_t ws_size,
                              hipStream_t stream) {
    const int*   ei    = (const int*)  d_in[0];   // edge_index [2, E]
    const float* ea    = (const float*)d_in[1];   // edge_attr  [E, 32]
    const int*   und   = (const int*)  d_in[2];   // undirected_mask [E]
    const float* Wall  = (const float*)d_in[3];   // [16, 32]
    const float* Wskip = (const float*)d_in[4];   // [16, 32]
    const float* bias  = (const float*)d_in[5];   // [16]
    float* out = (float*)d_out;                   // [E, 16]
    float* z   = (float*)d_ws;                    // [N, 16] node accumulator

    MEC_zero<<<(NUM_NODES * OUT_CH + 255) / 256, 256, 0, stream>>>(z);
    MEC_wmma_scatter<<<NUM_EDGES / 16 / 8, 256, 0, stream>>>(
        ei, ea, und, Wall, Wskip, bias, out, z);
    MEC_gather<<<(NUM_EDGES * OUT_CH + 255) / 256, 256, 0, stream>>>(ei, und, z, out);
}